// RankNetLoss_46093589021252
// MI455X (gfx1250) — compile-verified
//
#include <hip/hip_runtime.h>

// ---------------------------------------------------------------------------
// RankNet pairwise loss, n=8192, fp32.
//
// Roofline: 64 KB of input vs 67M virtual pairs -> compute bound on
// v_exp_f32 / v_log_f32 (TRANS). Symmetry trick processes each unordered
// pair once: C(i,j)+C(j,i) = |d|*log2e + 2*log2(1 + 2^(-|d|*log2e)),
// and the 0.5*(1-S)*d terms cancel exactly -> 2 TRANS per TWO ordered pairs.
// Diagonal 512x512 tiles use the faithful ordered formula (self pairs give
// exactly 1.0 and are subtracted in the finalize kernel).
//
// WMMA: V_WMMA_F32_16X16X4_F32 with an all-ones B matrix performs the
// intra-wave (32-lane) reduction of partial sums in one matrix op.
// ---------------------------------------------------------------------------

typedef __attribute__((ext_vector_type(2))) float v2f;
typedef __attribute__((ext_vector_type(8))) float v8f;

#define LOG2E_F 1.4426950408889634f
#define TILE 512
#define NT   256   // threads per block = 8 wave32
// TILE == 2*NT: each thread owns 2 rows.

// Wave-level sum of one float per lane using V_WMMA_F32_16X16X4_F32.
// A (16x4, 2 VGPRs): VGPR0 = partial, VGPR1 = 0. Per ISA layout, lanes 0-15
// land in A[m][0], lanes 16-31 in A[m][2], K=1/3 are zero. B = all ones
// (layout-invariant). D[m][n] = p(m) + p(m+16) for every n, so summing the
// 8 D VGPRs per lane gives sum(rows 0-7) in lanes 0-15 and sum(rows 8-15)
// in lanes 16-31; one xor-16 shuffle completes the 32-lane total.
__device__ __forceinline__ float wave_reduce_wmma(float x) {
    v2f a; a.x = x;    a.y = 0.0f;
    v2f b; b.x = 1.0f; b.y = 1.0f;
    v8f c = {};
    v8f d = __builtin_amdgcn_wmma_f32_16x16x4_f32(
        /*neg_a=*/false, a, /*neg_b=*/false, b,
        /*c_mod=*/(short)0, c, /*reuse_a=*/false, /*reuse_b=*/false);
    float h = ((d[0] + d[1]) + (d[2] + d[3])) + ((d[4] + d[5]) + (d[6] + d[7]));
    h += __shfl_xor(h, 16, 32);   // wave32: combine the two half-wave sums
    return h;                      // every lane holds the wave total
}

__global__ __launch_bounds__(NT) void ranknet_pair_sum(
    const float* __restrict__ score, const int* __restrict__ label,
    float* __restrict__ partials, int n)
{
    const int bi = blockIdx.y;           // row tile
    const int bj = blockIdx.x;           // col tile
    if (bj < bi) return;                 // lower triangle covered by symmetry

    __shared__ float2 tile[TILE];        // (score_j, label_j as float)
    __shared__ float  wsum[NT / 32];

    const int tid     = threadIdx.x;
    const int rowBase = bi * TILE;
    const int colBase = bj * TILE;

    // Stage the column tile into LDS (coalesced b64 stores, broadcast reads).
    for (int k = tid; k < TILE; k += NT) {
        const int c = colBase + k;
        tile[k] = make_float2(score[c], (float)label[c]);
    }

    // Two rows per thread, held in registers.
    const float s0 = score[rowBase + tid];
    const float s1 = score[rowBase + NT + tid];
    const float l0 = (float)label[rowBase + tid];        // labels in [0,5): exact in f32
    const float l1 = (float)label[rowBase + NT + tid];

    __syncthreads();

    float acc0 = 0.0f, acc1 = 0.0f;

    if (bi == bj) {
        // Diagonal tile: ordered formula with label term.
        // C = max(t,0) + log2(1 + 2^(-|t|))  with t = -d*log2e   (stable logaddexp)
        //   + 0.5*d if labels equal. Self pairs (i==j) contribute exactly 1.0.
        #pragma unroll 8
        for (int j = 0; j < TILE; ++j) {
            const float2 cj = tile[j];
            {
                float d   = s0 - cj.x;
                float t   = -d * LOG2E_F;
                float e   = __builtin_amdgcn_exp2f(-__builtin_fabsf(t)); // v_exp_f32, free |.| & neg mods
                float lg  = __builtin_amdgcn_logf(1.0f + e);             // v_log_f32 == log2
                float cc  = __builtin_fmaxf(t, 0.0f) + lg;
                float sel = (l0 == cj.y) ? d : 0.0f;                     // (1-S) term
                acc0 = __builtin_fmaf(0.5f, sel, acc0 + cc);
            }
            {
                float d   = s1 - cj.x;
                float t   = -d * LOG2E_F;
                float e   = __builtin_amdgcn_exp2f(-__builtin_fabsf(t));
                float lg  = __builtin_amdgcn_logf(1.0f + e);
                float cc  = __builtin_fmaxf(t, 0.0f) + lg;
                float sel = (l1 == cj.y) ? d : 0.0f;
                acc1 = __builtin_fmaf(0.5f, sel, acc1 + cc);
            }
        }
    } else {
        // Off-diagonal tile: each unordered pair once.
        // C(i,j)+C(j,i) = t + 2*log2(1 + 2^(-t)),  t = |d|*log2e.
        // Label terms cancel exactly -> no label reads, 2 TRANS / 2 ordered pairs.
        #pragma unroll 8
        for (int j = 0; j < TILE; ++j) {
            const float sj = tile[j].x;
            {
                float d  = s0 - sj;
                float t  = __builtin_fabsf(d) * LOG2E_F;
                float e  = __builtin_amdgcn_exp2f(-t);
                float lg = __builtin_amdgcn_logf(1.0f + e);
                acc0 += __builtin_fmaf(2.0f, lg, t);
            }
            {
                float d  = s1 - sj;
                float t  = __builtin_fabsf(d) * LOG2E_F;
                float e  = __builtin_amdgcn_exp2f(-t);
                float lg = __builtin_amdgcn_logf(1.0f + e);
                acc1 += __builtin_fmaf(2.0f, lg, t);
            }
        }
    }

    // Full EXEC here (uniform control flow) -> WMMA legal.
    const float wtot = wave_reduce_wmma(acc0 + acc1);
    if ((tid & 31) == 0) wsum[tid >> 5] = wtot;
    __syncthreads();
    if (tid == 0) {
        float b = 0.0f;
        #pragma unroll
        for (int w = 0; w < NT / 32; ++w) b += wsum[w];
        partials[bi * gridDim.x + bj] = b;   // upper-triangle entries only
    }
}

__global__ __launch_bounds__(256) void ranknet_finalize(
    const float* __restrict__ partials, int T, float* __restrict__ out, int n)
{
    __shared__ float red[256];
    const int tid = threadIdx.x;
    float v = 0.0f;
    for (int i = tid; i < T * T; i += 256) {
        const int r = i / T, c = i % T;
        if (c >= r) v += partials[i];        // only entries kernel 1 wrote
    }
    red[tid] = v;
    __syncthreads();
    for (int s = 128; s > 0; s >>= 1) {      // deterministic tree reduction
        if (tid < s) red[tid] += red[tid + s];
        __syncthreads();
    }
    if (tid == 0) {
        const float total  = red[0] - (float)n;            // drop i==j (each exactly 1.0)
        const float npairs = (float)n * (float)(n - 1);
        out[0] = total / npairs;
    }
}

extern "C" void kernel_launch(void* const* d_in, const int* in_sizes, int n_in,
                              void* d_out, int out_size, void* d_ws, size_t ws_size,
                              hipStream_t stream)
{
    const float* score = (const float*)d_in[0];
    const int*   label = (const int*)d_in[1];
    float*       out   = (float*)d_out;
    const int    n     = in_sizes[0];        // 8192
    const int    T     = n / TILE;           // 16 tiles per dim

    float* partials = (float*)d_ws;          // T*T floats of scratch

    dim3 grid(T, T);                         // lower triangle retires instantly
    ranknet_pair_sum<<<grid, NT, 0, stream>>>(score, label, partials, n);
    ranknet_finalize<<<1, 256, 0, stream>>>(partials, T, out, n);
}